// EdgeEnhancedMultiheadAttention_21492016349943
// MI455X (gfx1250) — compile-verified
//
#include <hip/hip_runtime.h>
#include <math.h>

#define D_MODEL 128
#define EDGE_DIM 32
#define N_HEADS 8
#define HEAD_DIM 16
#define INV_SCALE 0.25f   /* 1/sqrt(16) */
#define LN_EPS 1e-5f

typedef __attribute__((ext_vector_type(2))) float v2f;
typedef __attribute__((ext_vector_type(8))) float v8f;

__device__ __forceinline__ v8f wmma_f32_16x16x4(v2f a, v2f b, v8f c) {
    // D = A(16x4,f32) * B(4x16,f32) + C(16x16,f32)
    return __builtin_amdgcn_wmma_f32_16x16x4_f32(
        false, a, false, b, (short)0, c, false, false);
}

// ---------------------------------------------------------------------------
// One 16x128 projection: out tile = Xs(16x128) @ W(128x128) + b, this wave's
// 16-column slab. Separate inline function per matrix keeps the W pointer
// chained directly to the kernel argument -> global_load (not flat_load).
// ---------------------------------------------------------------------------
__device__ __forceinline__ void proj_tile_128(const float (&Xs)[16][132],
                                              const float* __restrict__ W,
                                              const float* __restrict__ b,
                                              float* __restrict__ O,
                                              int row0, int N, int lane, int wave)
{
    const int mrow  = lane & 15;
    const int khalf = (lane >> 4) * 2;
    const int col   = wave * 16 + (lane & 15);
    const int rbase = (lane >> 4) * 8;

    v8f acc = {0.f,0.f,0.f,0.f,0.f,0.f,0.f,0.f};
    #pragma unroll
    for (int kk = 0; kk < D_MODEL / 4; ++kk) {
        int kb = kk * 4 + khalf;
        v2f a, bb;
        a.x  = Xs[mrow][kb];
        a.y  = Xs[mrow][kb + 1];
        bb.x = W[(size_t)kb * D_MODEL + col];
        bb.y = W[(size_t)(kb + 1) * D_MODEL + col];
        acc = wmma_f32_16x16x4(a, bb, acc);
    }
    float bias = b[col];
    #pragma unroll
    for (int v = 0; v < 8; ++v) {
        int gr = row0 + rbase + v;
        if (gr < N) O[(size_t)gr * D_MODEL + col] = acc[v] + bias;
    }
}

// ---------------------------------------------------------------------------
// Kernel 1: Q = x@Wq+bq, K = x@Wk+bk, V = x@Wv+bv   (N x 128 each)
// block = 256 threads (8 waves), 16 rows per block, each wave owns 16 cols.
// ---------------------------------------------------------------------------
__global__ __launch_bounds__(256)
void qkv_proj_kernel(const float* __restrict__ x,
                     const float* __restrict__ Wq, const float* __restrict__ bq,
                     const float* __restrict__ Wk, const float* __restrict__ bk,
                     const float* __restrict__ Wv, const float* __restrict__ bv,
                     float* __restrict__ Q, float* __restrict__ K,
                     float* __restrict__ V, int N)
{
    __shared__ float Xs[16][132];   // stride 132: 16B-aligned rows, conflict-free

    const int tid  = threadIdx.x;
    const int lane = tid & 31;
    const int wave = tid >> 5;
    const int row0 = blockIdx.x * 16;

    #pragma unroll
    for (int i = 0; i < 8; ++i) {
        int e = tid + i * 256;
        int r = e >> 7, c = e & 127;
        int gr = row0 + r;
        Xs[r][c] = (gr < N) ? x[(size_t)gr * D_MODEL + c] : 0.0f;
    }
    __syncthreads();

    proj_tile_128(Xs, Wq, bq, Q, row0, N, lane, wave);
    proj_tile_128(Xs, Wk, bk, K, row0, N, lane, wave);
    proj_tile_128(Xs, Wv, bv, V, row0, N, lane, wave);
}

// ---------------------------------------------------------------------------
// Kernel 2: fused edge projection + attention + scatter (persistent blocks).
// block = 128 threads (4 waves); each wave owns 16 edges per tile iteration.
// Weq/Wev (+biases) staged in LDS once per block; edge_attr tiles staged with
// async global->LDS loads; e_q/e_v built with WMMA into per-wave LDS tiles.
// ---------------------------------------------------------------------------
__global__ __launch_bounds__(128)
void edge_attn_kernel(const float* __restrict__ edge_attr,
                      const int* __restrict__ edge_index,
                      const float* __restrict__ Weq, const float* __restrict__ beq,
                      const float* __restrict__ Wev, const float* __restrict__ bev,
                      const float* __restrict__ Q, const float* __restrict__ K,
                      const float* __restrict__ V,
                      float* __restrict__ acc_out, float* __restrict__ norm_out,
                      int E, int ntiles)
{
    __shared__ float Wqs[EDGE_DIM][136];  // stride 136: disjoint-bank B reads
    __shared__ float Wvs[EDGE_DIM][136];
    __shared__ float bqs[D_MODEL];
    __shared__ float bvs[D_MODEL];
    __shared__ float Es[4][16][36];       // edge_attr tile, stride 36 (144 B rows)
    __shared__ float Ts[4][16][132];      // per-wave 16x128 q (then v) tile

    const int tid  = threadIdx.x;
    const int lane = tid & 31;
    const int wave = tid >> 5;

    // ---- stage weights + biases once per block ----
    for (int i = tid; i < EDGE_DIM * D_MODEL; i += 128) {
        int r = i >> 7, c = i & 127;
        Wqs[r][c] = Weq[i];
        Wvs[r][c] = Wev[i];
    }
    if (tid < D_MODEL) {
        bqs[tid] = beq[tid];
        bvs[tid] = bev[tid];
    }

    const int mrow  = lane & 15;
    const int khalf = (lane >> 4) * 2;
    const int nlane = lane & 15;
    const int rbase = (lane >> 4) * 8;
    const int er    = lane & 15;       // edge row for scalar phases
    const int s     = lane >> 4;       // dim half (64s .. 64s+63)
    const int d0    = s * 64;

    const unsigned lds_es = (unsigned)(uintptr_t)&Es[wave][0][0];

    for (int tile = blockIdx.x; tile < ntiles; tile += gridDim.x) {
        const int e0 = tile * 64 + wave * 16;

        // prefetch next tile's edge_attr rows
        int ptile = tile + gridDim.x;
        if (ptile < ntiles) {
            const float* p = edge_attr +
                ((size_t)ptile * 64 + wave * 16 + (lane >> 1)) * EDGE_DIM +
                (size_t)(lane & 1) * 16;
            __builtin_prefetch(p, 0, 1);
        }

        // ---- stage edge_attr 16x32 tile (per wave), async fast path ----
        if (e0 + 16 <= E) {
            const char* gbase = (const char*)(edge_attr + (size_t)e0 * EDGE_DIM);
            #pragma unroll
            for (int inst = 0; inst < 4; ++inst) {
                int c   = lane + inst * 32;      // 128 chunks of 16 B
                int row = c >> 3, cc = c & 7;    // 8 chunks per 128 B row
                unsigned    ldsa = lds_es + row * 144 + cc * 16;
                const char* ga   = gbase + row * 128 + cc * 16;
                asm volatile("global_load_async_to_lds_b128 %0, %1, off"
                             :: "v"(ldsa), "v"(ga) : "memory");
            }
            asm volatile("s_wait_asynccnt 0" ::: "memory");
        } else {
            for (int i = lane; i < 16 * EDGE_DIM; i += 32) {
                int r = i >> 5, c = i & 31;
                int ge2 = e0 + r;
                Es[wave][r][c] = (ge2 < E) ? edge_attr[(size_t)ge2 * EDGE_DIM + c]
                                           : 0.0f;
            }
        }
        __syncthreads();

        // ---- e_q via WMMA (B from LDS), store (+bias) to Ts ----
        #pragma unroll
        for (int t = 0; t < 8; ++t) {
            const int tcol = t * 16 + nlane;
            v8f acc = {0.f,0.f,0.f,0.f,0.f,0.f,0.f,0.f};
            #pragma unroll
            for (int kk = 0; kk < EDGE_DIM / 4; ++kk) {
                int kb = kk * 4 + khalf;
                v2f a, b;
                a.x = Es[wave][mrow][kb];
                a.y = Es[wave][mrow][kb + 1];
                b.x = Wqs[kb][tcol];
                b.y = Wqs[kb + 1][tcol];
                acc = wmma_f32_16x16x4(a, b, acc);
            }
            float bias = bqs[tcol];
            #pragma unroll
            for (int v = 0; v < 8; ++v)
                Ts[wave][rbase + v][tcol] = acc[v] + bias;
        }
        __syncthreads();

        // ---- attention: 2 lanes per edge ----
        const int ge  = e0 + er;
        const int gec = (ge < E) ? ge : 0;
        const int src = edge_index[gec];
        const int dst = edge_index[(size_t)E + gec];

        const float* Qrow = Q + (size_t)src * D_MODEL + d0;
        const float* Krow = K + (size_t)dst * D_MODEL + d0;

        float sum_e = 0.f;
        #pragma unroll
        for (int hh = 0; hh < 4; ++hh) {
            float dot = 0.f;
            #pragma unroll
            for (int j = 0; j < HEAD_DIM; ++j) {
                int c = hh * 16 + j;
                float qv = Ts[wave][er][d0 + c] + Qrow[c];
                dot += qv * Krow[c];
            }
            float sc = dot * INV_SCALE;
            sc = fminf(fmaxf(sc, -5.0f), 5.0f);
            sum_e += __expf(sc);
        }
        sum_e += __shfl_xor(sum_e, 16, 32);
        const float attn = sum_e * (1.0f / (float)N_HEADS);
        if (ge < E && s == 0) atomicAdd(&norm_out[dst], attn);
        __syncthreads();   // q reads done before Ts is overwritten

        // ---- e_v via WMMA (B from LDS), store (+bias) to Ts (reuse) ----
        #pragma unroll
        for (int t = 0; t < 8; ++t) {
            const int tcol = t * 16 + nlane;
            v8f acc = {0.f,0.f,0.f,0.f,0.f,0.f,0.f,0.f};
            #pragma unroll
            for (int kk = 0; kk < EDGE_DIM / 4; ++kk) {
                int kb = kk * 4 + khalf;
                v2f a, b;
                a.x = Es[wave][mrow][kb];
                a.y = Es[wave][mrow][kb + 1];
                b.x = Wvs[kb][tcol];
                b.y = Wvs[kb + 1][tcol];
                acc = wmma_f32_16x16x4(a, b, acc);
            }
            float bias = bvs[tcol];
            #pragma unroll
            for (int v = 0; v < 8; ++v)
                Ts[wave][rbase + v][tcol] = acc[v] + bias;
        }
        __syncthreads();

        // ---- weighted scatter ----
        if (ge < E) {
            const float* Vrow = V + (size_t)src * D_MODEL + d0;
            float* arow = acc_out + (size_t)dst * D_MODEL + d0;
            #pragma unroll
            for (int c = 0; c < 64; ++c) {
                float vv = Ts[wave][er][d0 + c] + Vrow[c];
                atomicAdd(&arow[c], vv * attn);
            }
        }
        __syncthreads();   // scatter reads done before next tile's Ts writes
    }
}

// ---------------------------------------------------------------------------
// Kernel 3: out = (acc/(norm+1e-8)) @ Wo + bo + x ; LayerNorm(out).
// block = 256 threads (8 waves), 16 rows per block.
// ---------------------------------------------------------------------------
__global__ __launch_bounds__(256)
void out_ln_kernel(const float* __restrict__ acc_in,
                   const float* __restrict__ norm_in,
                   const float* __restrict__ x,
                   const float* __restrict__ Wo, const float* __restrict__ bo,
                   const float* __restrict__ gamma, const float* __restrict__ beta,
                   float* __restrict__ y, int N)
{
    __shared__ float As[16][132];
    __shared__ float Ys[16][132];
    __shared__ float inv_n[16];

    const int tid  = threadIdx.x;
    const int lane = tid & 31;
    const int wave = tid >> 5;
    const int row0 = blockIdx.x * 16;

    if (tid < 16) {
        int gr = row0 + tid;
        float nv = (gr < N) ? norm_in[gr] : 1.0f;
        inv_n[tid] = 1.0f / (nv + 1e-8f);
    }
    __syncthreads();

    #pragma unroll
    for (int i = 0; i < 8; ++i) {
        int e = tid + i * 256;
        int r = e >> 7, c = e & 127;
        int gr = row0 + r;
        As[r][c] = (gr < N) ? acc_in[(size_t)gr * D_MODEL + c] * inv_n[r] : 0.0f;
    }
    __syncthreads();

    const int mrow  = lane & 15;
    const int khalf = (lane >> 4) * 2;
    const int col   = wave * 16 + (lane & 15);
    const int rbase = (lane >> 4) * 8;

    v8f acc = {0.f,0.f,0.f,0.f,0.f,0.f,0.f,0.f};
    #pragma unroll
    for (int kk = 0; kk < D_MODEL / 4; ++kk) {
        int kb = kk * 4 + khalf;
        v2f a, b;
        a.x = As[mrow][kb];
        a.y = As[mrow][kb + 1];
        b.x = Wo[(size_t)kb * D_MODEL + col];
        b.y = Wo[(size_t)(kb + 1) * D_MODEL + col];
        acc = wmma_f32_16x16x4(a, b, acc);
    }
    float bias = bo[col];
    #pragma unroll
    for (int v = 0; v < 8; ++v) {
        int r = rbase + v;
        int gr = row0 + r;
        float resid = (gr < N) ? x[(size_t)gr * D_MODEL + col] : 0.0f;
        Ys[r][col] = acc[v] + bias + resid;
    }
    __syncthreads();

    // LayerNorm: wave w handles rows 2w, 2w+1; 32 lanes x 4 elems per row
    for (int rr = 0; rr < 2; ++rr) {
        int r  = wave * 2 + rr;
        int gr = row0 + r;
        float vals[4];
        float sum = 0.f, sq = 0.f;
        #pragma unroll
        for (int i = 0; i < 4; ++i) {
            float vv = Ys[r][lane * 4 + i];
            vals[i] = vv;
            sum += vv;
            sq  += vv * vv;
        }
        #pragma unroll
        for (int off = 16; off > 0; off >>= 1) {
            sum += __shfl_xor(sum, off, 32);
            sq  += __shfl_xor(sq,  off, 32);
        }
        float mu  = sum * (1.0f / (float)D_MODEL);
        float var = sq * (1.0f / (float)D_MODEL) - mu * mu;
        float rs  = rsqrtf(var + LN_EPS);
        if (gr < N) {
            #pragma unroll
            for (int i = 0; i < 4; ++i) {
                int d = lane * 4 + i;
                y[(size_t)gr * D_MODEL + d] = (vals[i] - mu) * rs * gamma[d] + beta[d];
            }
        }
    }
}

// ---------------------------------------------------------------------------
extern "C" void kernel_launch(void* const* d_in, const int* in_sizes, int n_in,
                              void* d_out, int out_size, void* d_ws, size_t ws_size,
                              hipStream_t stream)
{
    const float* x          = (const float*)d_in[0];
    const int*   edge_index = (const int*)  d_in[1];
    const float* edge_attr  = (const float*)d_in[2];
    const float* Wq  = (const float*)d_in[3];
    const float* bq  = (const float*)d_in[4];
    const float* Wk  = (const float*)d_in[5];
    const float* bk  = (const float*)d_in[6];
    const float* Wv  = (const float*)d_in[7];
    const float* bv  = (const float*)d_in[8];
    const float* Wo  = (const float*)d_in[9];
    const float* bo  = (const float*)d_in[10];
    const float* Weq = (const float*)d_in[11];
    const float* beq = (const float*)d_in[12];
    const float* Wev = (const float*)d_in[13];
    const float* bev = (const float*)d_in[14];
    const float* gamma = (const float*)d_in[15];
    const float* beta  = (const float*)d_in[16];
    float* y = (float*)d_out;

    const int N = in_sizes[0] / D_MODEL;
    const int E = in_sizes[2] / EDGE_DIM;

    float* Qb    = (float*)d_ws;
    float* Kb    = Qb + (size_t)N * D_MODEL;
    float* Vb    = Kb + (size_t)N * D_MODEL;
    float* accb  = Vb + (size_t)N * D_MODEL;
    float* normb = accb + (size_t)N * D_MODEL;

    // zero accumulator + norm (contiguous)
    hipMemsetAsync(accb, 0, ((size_t)N * D_MODEL + (size_t)N) * sizeof(float), stream);

    const int gRows = (N + 15) / 16;
    qkv_proj_kernel<<<gRows, 256, 0, stream>>>(x, Wq, bq, Wk, bk, Wv, bv,
                                               Qb, Kb, Vb, N);

    const int ntiles = (E + 63) / 64;
    int gEdges = ntiles < 1024 ? ntiles : 1024;
    edge_attn_kernel<<<gEdges, 128, 0, stream>>>(edge_attr, edge_index,
                                                 Weq, beq, Wev, bev,
                                                 Qb, Kb, Vb, accb, normb,
                                                 E, ntiles);

    out_ln_kernel<<<gRows, 256, 0, stream>>>(accb, normb, x, Wo, bo,
                                             gamma, beta, y, N);
}